// HashedEmbeddingBag_64742337020519
// MI455X (gfx1250) — compile-verified
//
#include <hip/hip_runtime.h>

// ---------------------------------------------------------------------------
// Hashed embedding gather for MI455X (gfx1250, wave32).
//
// out[b,h,d] = W[ ((a*rn1 + d*rn2 + rn3) mod rn0) mod WS ],  a = indices[b,h]
//
// Memory-bound kernel: 210 MB non-temporal stores (HBM floor ~9 us @23.3TB/s),
// gathers hit the L2-resident 4 MB table. One wave per (b,h) pair; lanes span
// the 64-wide embedding dim, 2 floats per lane -> coalesced 256B b64 stores.
// ---------------------------------------------------------------------------

typedef __attribute__((ext_vector_type(2))) float v2f;

// Structural constants of the reference (rn[0] is hardcoded in the reference;
// WEIGHT_SIZE = 1e6*64/64 + 1). Compile-time divisors -> magic-mul division,
// no runtime 64-bit urem.
constexpr unsigned long long RN0   = 2038074743ull;   // rn[0], < 2^31
constexpr unsigned int       RN0_32 = 2038074743u;
constexpr unsigned int       WS    = 1000001u;        // compressed table size

constexpr int PAIRS_PER_WAVE  = 8;
constexpr int WAVES_PER_BLOCK = 8;    // 256 threads = 8 wave32
constexpr int PAIRS_PER_BLOCK = PAIRS_PER_WAVE * WAVES_PER_BLOCK;   // 64

__global__ __launch_bounds__(256) void hashed_embedding_gather(
    const long long* __restrict__ indices,   // [B*H] int64, values < 1e6
    const float*     __restrict__ weight,    // [WS] f32 (4 MB, L2-resident)
    const long long* __restrict__ rn,        // [51] int64
    float*           __restrict__ out,       // [B*H*64] f32
    int num_pairs)
{
  const unsigned lane = threadIdx.x & 31u;
  const unsigned wave = threadIdx.x >> 5;

  // rn1/rn2/rn3: uniform addresses -> scalar loads.
  const unsigned long long rn1 = (unsigned long long)rn[1];
  const unsigned long long rn2 = (unsigned long long)rn[2];
  const unsigned long long rn3 = (unsigned long long)rn[3];

  // Per-lane d contribution, computed once: dr_d = (d * rn2) mod rn0 < 2^31.
  const unsigned d0  = lane << 1;                       // lane handles d0, d0+1
  const unsigned dr0 = (unsigned)(((unsigned long long)d0       * rn2) % RN0);
  const unsigned dr1 = (unsigned)(((unsigned long long)(d0 + 1) * rn2) % RN0);

  // Wave-uniform pair base; readfirstlane pins it to an SGPR so the index
  // fetch below becomes s_load_b64 and the per-pair hash rides the SALU.
  unsigned pairBase = (blockIdx.x * WAVES_PER_BLOCK + wave) * PAIRS_PER_WAVE;
  pairBase = __builtin_amdgcn_readfirstlane(pairBase);

#pragma unroll
  for (int i = 0; i < PAIRS_PER_WAVE; ++i) {
    const unsigned p = pairBase + (unsigned)i;
    if (p >= (unsigned)num_pairs) break;

    // a < 1e6 < rn0, so (a mod rn0) == a and a*rn1 + rn3 < 2^62: one 64-bit
    // magic-mod by the constant rn0, once per 64 output elements.
    const unsigned long long a = (unsigned long long)indices[p];   // s_load
    const unsigned c = (unsigned)((a * rn1 + rn3) % RN0);          // < 2^31

    // Per-element: u32 add + conditional subtract (both operands < 2^31,
    // sum < 2^32), then 32-bit magic-mod by WS. ~8 VALU ops/element.
    unsigned h0 = c + dr0; if (h0 >= RN0_32) h0 -= RN0_32;
    unsigned h1 = c + dr1; if (h1 >= RN0_32) h1 -= RN0_32;

    v2f v;
    v.x = weight[h0 % WS];      // random gather, L2/WGP$ hit
    v.y = weight[h1 % WS];

    // Output is write-once & never re-read: non-temporal store keeps the hot
    // weight table resident in WGP$/L2 instead of streaming stores evicting it.
    v2f* dst = (v2f*)(out + (size_t)p * 64u + d0);   // 8B aligned, 256B/wave
    __builtin_nontemporal_store(v, dst);
  }
}

extern "C" void kernel_launch(void* const* d_in, const int* in_sizes, int n_in,
                              void* d_out, int out_size, void* d_ws, size_t ws_size,
                              hipStream_t stream) {
  (void)n_in; (void)out_size; (void)d_ws; (void)ws_size;

  const long long* indices = (const long long*)d_in[0];  // int64 [4096*200]
  const float*     weight  = (const float*)d_in[1];      // f32   [1000001]
  const long long* rn      = (const long long*)d_in[2];  // int64 [51]
  float*           out     = (float*)d_out;              // f32   [4096*200*64]

  const int num_pairs = in_sizes[0];                     // 819200
  const int grid = (num_pairs + PAIRS_PER_BLOCK - 1) / PAIRS_PER_BLOCK;

  hipLaunchKernelGGL(hashed_embedding_gather, dim3(grid), dim3(256), 0, stream,
                     indices, weight, rn, out, num_pairs);
}